// SmartPool2d_46566035423962
// MI455X (gfx1250) — compile-verified
//
#include <hip/hip_runtime.h>

typedef __attribute__((ext_vector_type(2))) float v2f;
typedef __attribute__((ext_vector_type(8))) float v8f;

#define H 128
#define W 128
#define OUT 64
#define KT 7           // 2*SCALE+3 taps
#define RATIO 0.1f
#define XP 132         // padded stride for 128-wide X rows (16B aligned rows)
#define WP 132         // Wr stride: [64][132]
#define YP 132         // Y stride:  [64][132]
#define CP 66          // WcT stride: [128][66] (8448 floats, reuses Wr buffer)

__global__ __launch_bounds__(256) void smartpool_kernel(const float* __restrict__ x,
                                                        float* __restrict__ out) {
    __shared__ float xs[H * XP];       // 128x128 channel tile (padded)
    __shared__ float wbuf[64 * WP];    // Wr [64][132]  /  reused as WcT [128][66]
    __shared__ float ys[64 * YP];      // intermediate Y [64][128] (padded)
    __shared__ float red[256];
    __shared__ int   rflag[H];
    __shared__ int   cflag[W];
    __shared__ int   bb[4];            // r0, r1, c0, c1

    const int t = threadIdx.x;
    const int ch = blockIdx.x;                       // 0 .. N*C-1
    const float* xc = x + (size_t)ch * (H * W);
    float* oc = out + (size_t)ch * (OUT * OUT);

    // ---- Phase 1: load channel into LDS (float4), running max ----
    float lmax = -3.402823466e38f;
    #pragma unroll
    for (int i = 0; i < 16; ++i) {
        int f4 = t + i * 256;                        // 0..4095
        float4 v = reinterpret_cast<const float4*>(xc)[f4];
        int e = f4 * 4;
        int r = e >> 7, c = e & 127;                 // row, col (col % 4 == 0)
        *reinterpret_cast<float4*>(&xs[r * XP + c]) = v;
        lmax = fmaxf(lmax, fmaxf(fmaxf(v.x, v.y), fmaxf(v.z, v.w)));
    }
    red[t] = lmax;
    __syncthreads();
    for (int s = 128; s > 0; s >>= 1) {
        if (t < s) red[t] = fmaxf(red[t], red[t + s]);
        __syncthreads();
    }
    const float thr = RATIO * red[0];

    // ---- Phase 2: row/col "any >= thr" flags, then bbox ----
    if (t < H) {
        int any = 0;
        for (int j = 0; j < W; ++j) any |= (xs[t * XP + j] >= thr);
        rflag[t] = any;
    } else {
        int c = t - H;
        int any = 0;
        for (int j = 0; j < H; ++j) any |= (xs[j * XP + c] >= thr);
        cflag[c] = any;
    }
    __syncthreads();
    if (t == 0) {
        int r0 = 0, r1 = H - 1, c0 = 0, c1 = W - 1;
        for (int j = 0; j < H; ++j)       { if (rflag[j]) { r0 = j; break; } }
        for (int j = H - 1; j >= 0; --j)  { if (rflag[j]) { r1 = j; break; } }
        for (int j = 0; j < W; ++j)       { if (cflag[j]) { c0 = j; break; } }
        for (int j = W - 1; j >= 0; --j)  { if (cflag[j]) { c1 = j; break; } }
        bb[0] = r0; bb[1] = r1; bb[2] = c0; bb[3] = c1;
    }
    __syncthreads();

    // ---- Phase 3: build Wr [64 x 128] (dense, normalized triangle filter) ----
    #pragma unroll
    for (int i = 0; i < 33; ++i) wbuf[t + i * 256] = 0.0f;   // 33*256 == 64*WP
    __syncthreads();
    if (t < OUT) {
        const int r0 = bb[0], r1 = bb[1];
        float s    = (float)(r1 - r0 + 1) / (float)OUT;
        float supp = fmaxf(s, 1.0f);
        float ctr  = (float)r0 + ((float)t + 0.5f) * s;
        int j0 = (int)floorf(ctr - supp + 0.5f);
        float wv[KT], wsum = 0.0f;
        #pragma unroll
        for (int k = 0; k < KT; ++k) {
            int j = j0 + k;
            float tt = ((float)j + 0.5f - ctr) / supp;
            float w  = fmaxf(0.0f, 1.0f - fabsf(tt));
            if (j < r0 || j > r1) w = 0.0f;
            wv[k] = w; wsum += w;
        }
        float inv = 1.0f / wsum;
        #pragma unroll
        for (int k = 0; k < KT; ++k) {
            int j = j0 + k;
            if (j >= r0 && j <= r1) wbuf[t * WP + j] = wv[k] * inv;
        }
    }
    __syncthreads();

    // ---- WMMA lane mapping (f32 16x16x4) ----
    const int wave = t >> 5;
    const int lane = t & 31;
    const int ma = lane & 15;            // A row within tile
    const int kb = (lane >> 4) * 2;      // A/B k-pair base (0 or 2)
    const int nn = lane & 15;            // B/C column within tile
    const int rb = (lane >> 4) << 3;     // C row offset (0 or 8)

    // ---- Phase 4: Y(64x128) = Wr(64x128) * X(128x128) ----
    #pragma unroll
    for (int q = 0; q < 4; ++q) {
        int tile = wave * 4 + q;         // 0..31
        int Mt = tile >> 3, Nt = tile & 7;
        v8f acc = {0.f, 0.f, 0.f, 0.f, 0.f, 0.f, 0.f, 0.f};
        for (int kk = 0; kk < 32; ++kk) {
            int k0 = kk * 4 + kb;
            v2f a, b;
            a.x = wbuf[(Mt * 16 + ma) * WP + k0];
            a.y = wbuf[(Mt * 16 + ma) * WP + k0 + 1];
            b.x = xs[k0 * XP + Nt * 16 + nn];
            b.y = xs[(k0 + 1) * XP + Nt * 16 + nn];
            acc = __builtin_amdgcn_wmma_f32_16x16x4_f32(false, a, false, b,
                                                        (short)0, acc, false, false);
        }
        int rbase = Mt * 16 + rb;
        int cc = Nt * 16 + nn;
        #pragma unroll
        for (int v = 0; v < 8; ++v) ys[(rbase + v) * YP + cc] = acc[v];
    }
    __syncthreads();

    // ---- Phase 5: build WcT [128 x 64] in wbuf (stride CP) ----
    #pragma unroll
    for (int i = 0; i < 33; ++i) wbuf[t + i * 256] = 0.0f;   // 33*256 == 128*CP
    __syncthreads();
    if (t < OUT) {
        const int c0 = bb[2], c1 = bb[3];
        float s    = (float)(c1 - c0 + 1) / (float)OUT;
        float supp = fmaxf(s, 1.0f);
        float ctr  = (float)c0 + ((float)t + 0.5f) * s;
        int j0 = (int)floorf(ctr - supp + 0.5f);
        float wv[KT], wsum = 0.0f;
        #pragma unroll
        for (int k = 0; k < KT; ++k) {
            int j = j0 + k;
            float tt = ((float)j + 0.5f - ctr) / supp;
            float w  = fmaxf(0.0f, 1.0f - fabsf(tt));
            if (j < c0 || j > c1) w = 0.0f;
            wv[k] = w; wsum += w;
        }
        float inv = 1.0f / wsum;
        #pragma unroll
        for (int k = 0; k < KT; ++k) {
            int j = j0 + k;
            if (j >= c0 && j <= c1) wbuf[j * CP + t] = wv[k] * inv;   // transposed
        }
    }
    __syncthreads();

    // ---- Phase 6: Z(64x64) = Y(64x128) * WcT(128x64), write to global ----
    #pragma unroll
    for (int q = 0; q < 2; ++q) {
        int tile = wave * 2 + q;         // 0..15
        int Mt = tile >> 2, Nt = tile & 3;
        v8f acc = {0.f, 0.f, 0.f, 0.f, 0.f, 0.f, 0.f, 0.f};
        for (int kk = 0; kk < 32; ++kk) {
            int k0 = kk * 4 + kb;
            v2f a, b;
            a.x = ys[(Mt * 16 + ma) * YP + k0];
            a.y = ys[(Mt * 16 + ma) * YP + k0 + 1];
            b.x = wbuf[k0 * CP + Nt * 16 + nn];
            b.y = wbuf[(k0 + 1) * CP + Nt * 16 + nn];
            acc = __builtin_amdgcn_wmma_f32_16x16x4_f32(false, a, false, b,
                                                        (short)0, acc, false, false);
        }
        int rbase = Mt * 16 + rb;
        int cc = Nt * 16 + nn;
        #pragma unroll
        for (int v = 0; v < 8; ++v) oc[(rbase + v) * OUT + cc] = acc[v];
    }
}

extern "C" void kernel_launch(void* const* d_in, const int* in_sizes, int n_in,
                              void* d_out, int out_size, void* d_ws, size_t ws_size,
                              hipStream_t stream) {
    const float* x = (const float*)d_in[0];
    float* out = (float*)d_out;
    int channels = in_sizes[0] / (H * W);            // N*C = 1024
    smartpool_kernel<<<dim3(channels), dim3(256), 0, stream>>>(x, out);
}